// AntiCommunityGNN_21010980012300
// MI455X (gfx1250) — compile-verified
//
#include <hip/hip_runtime.h>
#include <hip/hip_bf16.h>
#include <stdint.h>

typedef __attribute__((ext_vector_type(2))) float v2f;
typedef __attribute__((ext_vector_type(8))) float v8f;

// ---------------------------------------------------------------------------
// Zero a float buffer
// ---------------------------------------------------------------------------
__global__ __launch_bounds__(256) void zero_kernel(float* __restrict__ p, int n) {
    int i = blockIdx.x * blockDim.x + threadIdx.x;
    if (i < n) p[i] = 0.0f;
}

// ---------------------------------------------------------------------------
// Native f32 L2 atomic add (no return -> STOREcnt path, global_atomic_add_f32)
// ---------------------------------------------------------------------------
__device__ __forceinline__ void atomic_add_f32(float* p, float v) {
    __hip_atomic_fetch_add(p, v, __ATOMIC_RELAXED, __HIP_MEMORY_SCOPE_AGENT);
}

// Streaming (non-temporal) loads for the one-touch edge arrays: keep the
// hot per-node arrays (dinv/h/agg) resident in the 192MB L2 instead.
__device__ __forceinline__ int   nt_load_i32(const int* p)   { return __builtin_nontemporal_load(p); }
__device__ __forceinline__ float nt_load_f32(const float* p) { return __builtin_nontemporal_load(p); }

// ---------------------------------------------------------------------------
// deg[dst] += w   (one thread per edge)
// ---------------------------------------------------------------------------
__global__ __launch_bounds__(256) void degree_kernel(const int* __restrict__ ei,
                                                     const float* __restrict__ w,
                                                     float* __restrict__ deg, int e) {
    int t = blockIdx.x * blockDim.x + threadIdx.x;
    if (t < e) {
        int d = nt_load_i32(ei + e + t);     // edge_index row 1 = dst
        atomic_add_f32(&deg[d], nt_load_f32(w + t));
    }
}

// dinv[i] = rsqrt(deg[i] + 1)  (self-loop weight 1 folded in; deg+1 >= 1 > 0)
__global__ __launch_bounds__(256) void dinv_kernel(float* __restrict__ deg, int n) {
    int i = blockIdx.x * blockDim.x + threadIdx.x;
    if (i < n) deg[i] = rsqrtf(deg[i] + 1.0f);
}

// ---------------------------------------------------------------------------
// h1 = x @ W1   [n,2] x [2,8] -> [n,8]  via V_WMMA_F32_16X16X4_F32 (K padded)
// One wave per 16 nodes. A: 16x4 (K=2 real, K=2,3 zero). B: 4x16 (N=8 real).
// ---------------------------------------------------------------------------
__global__ __launch_bounds__(32) void linear1_wmma(const float* __restrict__ x,
                                                   const float* __restrict__ W1,
                                                   float* __restrict__ h1, int n) {
    int lane = threadIdx.x;          // 0..31
    int base = blockIdx.x * 16;
    int idx  = lane & 15;
    bool lo  = lane < 16;            // lanes 16-31 carry K=2,3 (zero pad)
    int r    = base + idx;
    bool rv  = (r < n);

    v2f a;
    a.x = (lo && rv) ? x[r * 2 + 0] : 0.0f;   // A[M=r][K=0]  (hi lanes: K=2 -> 0)
    a.y = (lo && rv) ? x[r * 2 + 1] : 0.0f;   // A[M=r][K=1]  (hi lanes: K=3 -> 0)

    bool cv = (idx < 8);
    v2f b;
    b.x = (lo && cv) ? W1[0 * 8 + idx] : 0.0f;  // B[K=0][N]  (hi lanes: K=2 -> 0)
    b.y = (lo && cv) ? W1[1 * 8 + idx] : 0.0f;  // B[K=1][N]  (hi lanes: K=3 -> 0)

    v8f c = {};
    v8f d = __builtin_amdgcn_wmma_f32_16x16x4_f32(false, a, false, b,
                                                  (short)0, c, false, false);
    // D layout: VGPR v holds row base+v (lanes 0-15) / base+v+8 (lanes 16-31), col = idx
    int rowoff = (lo ? 0 : 8) + base;
    if (base + 16 <= n) {            // uniform fast path: full tile, no row guards
        if (cv) {
#pragma unroll
            for (int v = 0; v < 8; ++v) h1[(rowoff + v) * 8 + idx] = d[v];
        }
    } else {
        if (cv) {
#pragma unroll
            for (int v = 0; v < 8; ++v) {
                int m = rowoff + v;
                if (m < n) h1[m * 8 + idx] = d[v];
            }
        }
    }
}

// ---------------------------------------------------------------------------
// h2 = h1 @ W2   [n,8] x [8,2] -> [n,2]  via two chained 16x16x4 WMMAs (K=8)
// ---------------------------------------------------------------------------
__global__ __launch_bounds__(32) void linear2_wmma(const float* __restrict__ h1,
                                                   const float* __restrict__ W2,
                                                   float* __restrict__ h2, int n) {
    int lane = threadIdx.x;
    int base = blockIdx.x * 16;
    int idx  = lane & 15;
    bool lo  = lane < 16;
    int r    = base + idx;
    bool rv  = (r < n);
    int k0   = lo ? 0 : 2;           // VGPR0 carries K=k0, VGPR1 carries K=k0+1

    v2f a0, a1;
    a0.x = rv ? h1[r * 8 + k0 + 0] : 0.0f;   // WMMA#1: K = 0..3
    a0.y = rv ? h1[r * 8 + k0 + 1] : 0.0f;
    a1.x = rv ? h1[r * 8 + k0 + 4] : 0.0f;   // WMMA#2: K = 4..7
    a1.y = rv ? h1[r * 8 + k0 + 5] : 0.0f;

    bool cv = (idx < 2);
    v2f b0, b1;
    b0.x = cv ? W2[(k0 + 0) * 2 + idx] : 0.0f;
    b0.y = cv ? W2[(k0 + 1) * 2 + idx] : 0.0f;
    b1.x = cv ? W2[(k0 + 4) * 2 + idx] : 0.0f;
    b1.y = cv ? W2[(k0 + 5) * 2 + idx] : 0.0f;

    v8f c = {};
    c = __builtin_amdgcn_wmma_f32_16x16x4_f32(false, a0, false, b0, (short)0, c, false, false);
    c = __builtin_amdgcn_wmma_f32_16x16x4_f32(false, a1, false, b1, (short)0, c, false, false);

    int rowoff = (lo ? 0 : 8) + base;
    if (base + 16 <= n) {            // uniform fast path
        if (cv) {
#pragma unroll
            for (int v = 0; v < 8; ++v) h2[(rowoff + v) * 2 + idx] = c[v];
        }
    } else {
        if (cv) {
#pragma unroll
            for (int v = 0; v < 8; ++v) {
                int m = rowoff + v;
                if (m < n) h2[m * 2 + idx] = c[v];
            }
        }
    }
}

// ---------------------------------------------------------------------------
// Edge aggregation, 8 features: agg[dst] += dinv[src]*w*dinv[dst] * h[src]
// ---------------------------------------------------------------------------
__global__ __launch_bounds__(256) void edge_agg8(const int* __restrict__ ei,
                                                 const float* __restrict__ w,
                                                 const float* __restrict__ dinv,
                                                 const float* __restrict__ h,
                                                 float* __restrict__ agg, int e) {
    int t = blockIdx.x * blockDim.x + threadIdx.x;
    if (t >= e) return;
    int s = nt_load_i32(ei + t);
    int d = nt_load_i32(ei + e + t);
    float nrm = dinv[s] * nt_load_f32(w + t) * dinv[d];
    const float4* hp = (const float4*)(h + (size_t)s * 8);
    float4 lo4 = hp[0];
    float4 hi4 = hp[1];
    float* ap = agg + (size_t)d * 8;
    atomic_add_f32(ap + 0, nrm * lo4.x);
    atomic_add_f32(ap + 1, nrm * lo4.y);
    atomic_add_f32(ap + 2, nrm * lo4.z);
    atomic_add_f32(ap + 3, nrm * lo4.w);
    atomic_add_f32(ap + 4, nrm * hi4.x);
    atomic_add_f32(ap + 5, nrm * hi4.y);
    atomic_add_f32(ap + 6, nrm * hi4.z);
    atomic_add_f32(ap + 7, nrm * hi4.w);
}

// Edge aggregation, 2 features
__global__ __launch_bounds__(256) void edge_agg2(const int* __restrict__ ei,
                                                 const float* __restrict__ w,
                                                 const float* __restrict__ dinv,
                                                 const float* __restrict__ h,
                                                 float* __restrict__ agg, int e) {
    int t = blockIdx.x * blockDim.x + threadIdx.x;
    if (t >= e) return;
    int s = nt_load_i32(ei + t);
    int d = nt_load_i32(ei + e + t);
    float nrm = dinv[s] * nt_load_f32(w + t) * dinv[d];
    const float2* hp = (const float2*)(h + (size_t)s * 2);
    float2 hv = hp[0];
    float* ap = agg + (size_t)d * 2;
    atomic_add_f32(ap + 0, nrm * hv.x);
    atomic_add_f32(ap + 1, nrm * hv.y);
}

// ---------------------------------------------------------------------------
// finalize1: h1[i] = relu(agg1[i] + dinv[i]^2 * h1[i] + b1)   (in-place)
// ---------------------------------------------------------------------------
__global__ __launch_bounds__(256) void finalize1(float* __restrict__ h1,
                                                 const float* __restrict__ agg1,
                                                 const float* __restrict__ dinv,
                                                 const float* __restrict__ b1, int n) {
    int i = blockIdx.x * blockDim.x + threadIdx.x;
    if (i >= n) return;
    float sl = dinv[i] * dinv[i];
#pragma unroll
    for (int k = 0; k < 8; ++k) {
        float v = agg1[(size_t)i * 8 + k] + sl * h1[(size_t)i * 8 + k] + b1[k];
        h1[(size_t)i * 8 + k] = v > 0.0f ? v : 0.0f;
    }
}

// finalize2 + 2-way softmax -> out
__global__ __launch_bounds__(256) void finalize2(const float* __restrict__ h2,
                                                 const float* __restrict__ agg2,
                                                 const float* __restrict__ dinv,
                                                 const float* __restrict__ b2,
                                                 float* __restrict__ out, int n) {
    int i = blockIdx.x * blockDim.x + threadIdx.x;
    if (i >= n) return;
    float sl = dinv[i] * dinv[i];
    float v0 = agg2[(size_t)i * 2 + 0] + sl * h2[(size_t)i * 2 + 0] + b2[0];
    float v1 = agg2[(size_t)i * 2 + 1] + sl * h2[(size_t)i * 2 + 1] + b2[1];
    float m  = fmaxf(v0, v1);
    float e0 = __expf(v0 - m);
    float e1 = __expf(v1 - m);
    float inv = 1.0f / (e0 + e1);
    out[(size_t)i * 2 + 0] = e0 * inv;
    out[(size_t)i * 2 + 1] = e1 * inv;
}

// ---------------------------------------------------------------------------
// Host launcher
// ---------------------------------------------------------------------------
static inline float* align16(float* p) {
    return (float*)(((uintptr_t)p + 15) & ~(uintptr_t)15);
}

extern "C" void kernel_launch(void* const* d_in, const int* in_sizes, int n_in,
                              void* d_out, int out_size, void* d_ws, size_t ws_size,
                              hipStream_t stream) {
    const float* x  = (const float*)d_in[0];
    const int*   ei = (const int*)d_in[1];   // [2,E] int32 (JAX canonicalizes int64)
    const float* w  = (const float*)d_in[2];
    const float* W1 = (const float*)d_in[3];
    const float* b1 = (const float*)d_in[4];
    const float* W2 = (const float*)d_in[5];
    const float* b2 = (const float*)d_in[6];
    float* out = (float*)d_out;

    const int n = in_sizes[0] / 2;   // nodes
    const int e = in_sizes[2];       // edges

    // Workspace layout (16B-aligned chunks):
    float* ws   = (float*)d_ws;
    float* dinv = align16(ws);                       //  n   floats (deg -> dinv)
    float* agg1 = align16(dinv + (size_t)n);         //  8n  floats
    float* agg2 = align16(agg1 + (size_t)n * 8);     //  2n  floats
    float* h1   = align16(agg2 + (size_t)n * 2);     //  8n  floats
    float* h2   = align16(h1   + (size_t)n * 8);     //  2n  floats

    const int TB = 256;
    dim3 blkE((e + TB - 1) / TB), thr(TB);
    dim3 blkN((n + TB - 1) / TB);
    dim3 blkN8((n * 8 + TB - 1) / TB);
    dim3 blkN2((n * 2 + TB - 1) / TB);
    dim3 blkW((n + 15) / 16), thrW(32);

    // 1) zero accumulators
    zero_kernel<<<blkN,  thr, 0, stream>>>(dinv, n);
    zero_kernel<<<blkN8, thr, 0, stream>>>(agg1, n * 8);
    zero_kernel<<<blkN2, thr, 0, stream>>>(agg2, n * 2);

    // 2) gcn_norm: degree + dinv (self-loop folded as +1)
    degree_kernel<<<blkE, thr, 0, stream>>>(ei, w, dinv, e);
    dinv_kernel<<<blkN, thr, 0, stream>>>(dinv, n);

    // 3) layer 1
    linear1_wmma<<<blkW, thrW, 0, stream>>>(x, W1, h1, n);
    edge_agg8<<<blkE, thr, 0, stream>>>(ei, w, dinv, h1, agg1, e);
    finalize1<<<blkN, thr, 0, stream>>>(h1, agg1, dinv, b1, n);

    // 4) layer 2 + softmax
    linear2_wmma<<<blkW, thrW, 0, stream>>>(h1, W2, h2, n);
    edge_agg2<<<blkE, thr, 0, stream>>>(ei, w, dinv, h2, agg2, e);
    finalize2<<<blkN, thr, 0, stream>>>(h2, agg2, dinv, b2, out, n);
}